// Segmenter_91207925498441
// MI455X (gfx1250) — compile-verified
//
#include <hip/hip_runtime.h>
#include <hip/hip_bf16.h>
#include <math.h>

// ---------------------------------------------------------------------------
// neuralef loss, MI455X (gfx1250, wave32, WMMA).
// Dominant op: fused fp32 Gram (FFn @ FFn^T, 103 GFLOP) + per-row top-10,
// using V_WMMA_F32_16X16X4_F32 so top-k index selection matches the f32 ref.
// ---------------------------------------------------------------------------

typedef __attribute__((ext_vector_type(2))) float v2f;
typedef __attribute__((ext_vector_type(4))) float v4f;
typedef __attribute__((ext_vector_type(8))) float v8f;

#define NTOK   8192
#define DFEAT  768
#define KEIG   256
#define KNN    10
#define LDA    772   // padded LDS row stride (772 % 64 == 4 -> conflict-free b64)
#define LDC    20    // C-tile row stride: 16B-aligned rows, conflict-free b128

// ---------------------------------------------------------------------------
// Kernel 1: row L2-normalize FF -> FFn; sq[i] = sum(FFn[i]^2)  (one wave/row)
// ---------------------------------------------------------------------------
__global__ __launch_bounds__(256) void normalize_kernel(
    const float* __restrict__ FF, float* __restrict__ FFn, float* __restrict__ sq) {
  const int wave = threadIdx.x >> 5, lane = threadIdx.x & 31;
  const int row  = blockIdx.x * 8 + wave;
  const float* src = FF  + (size_t)row * DFEAT;
  float*       dst = FFn + (size_t)row * DFEAT;
  float s = 0.f;
  for (int c = lane; c < DFEAT; c += 32) { float v = src[c]; s += v * v; }
#pragma unroll
  for (int o = 16; o > 0; o >>= 1) s += __shfl_xor(s, o, 32);
  const float inv = 1.0f / fmaxf(sqrtf(s), 1e-12f);
  float q = 0.f;
  for (int c = lane; c < DFEAT; c += 32) { float v = src[c] * inv; dst[c] = v; q += v * v; }
#pragma unroll
  for (int o = 16; o > 0; o >>= 1) q += __shfl_xor(q, o, 32);
  if (lane == 0) sq[row] = q;
}

// ---------------------------------------------------------------------------
// Kernel 2: fused similarity + per-row top-10.
// Block = 16 rows; 8 waves; wave w sweeps column tiles w, w+8, ...
// t = 2*dot - sq_j is formed while spilling the accumulator (sq load hidden
// behind the WMMA loop); scan reads t 4-at-a-time via ds_load_b128 with a
// vectorized max-guard, so the common path has no global loads and no stalls.
// ---------------------------------------------------------------------------
__global__ __launch_bounds__(256) void sim_topk_kernel(
    const float* __restrict__ FFn, const float* __restrict__ sq,
    float* __restrict__ tkv, int* __restrict__ tki) {
  __shared__ float sA[16 * LDA];          // 49408 B : A panel (16 x 768, padded)
  __shared__ float sC[8 * 16 * LDC];      // 10240 B : per-wave t tile / merge vals
  __shared__ int   sMI[8 * 16 * KNN];     //  5120 B : merge indices

  const int rowBase = blockIdx.x * 16;
  const int tid  = threadIdx.x;
  const int wave = tid >> 5, lane = tid & 31;
  const int lh   = lane >> 4;             // lane half (0/1) -> K sub-pair
  const int ll   = lane & 15;             // M (A) / N (B,C) index

  // stage the 16-row A panel into LDS
  for (int idx = tid; idx < 16 * DFEAT; idx += 256) {
    int r = idx / DFEAT, c = idx - r * DFEAT;
    sA[r * LDA + c] = FFn[(size_t)(rowBase + r) * DFEAT + c];
  }
  __syncthreads();

  float kv[KNN]; int ki[KNN];             // per-lane (row = rowBase+ll) top-10 on t
#pragma unroll
  for (int s = 0; s < KNN; ++s) { kv[s] = -3.0e38f; ki[s] = 0x7fffffff; }

  const float* aBase = &sA[ll * LDA + 2 * lh];
  float* ct = &sC[wave * (16 * LDC)];

  for (int tile = wave; tile < NTOK / 16; tile += 8) {
    const int colBase = tile * 16;
    const float sqc = sq[colBase + ll];   // issued early; latency hidden by GEMM
    const float* bBase = &FFn[(size_t)(colBase + ll) * DFEAT + 2 * lh];
    if (tile + 8 < NTOK / 16)             // prefetch next B panel (global_prefetch_b8)
      __builtin_prefetch(&FFn[(size_t)(colBase + 128 + ll) * DFEAT], 0, 0);

    v8f c = {};
#pragma unroll 4
    for (int k0 = 0; k0 < DFEAT; k0 += 4) {
      v2f a = *(const v2f*)(aBase + k0);                 // ds_load_2addr_b64
      v2f b = *(const v2f*)(bBase + k0);                 // global_load_b64
      c = __builtin_amdgcn_wmma_f32_16x16x4_f32(false, a, false, b,
                                                (short)0, c, false, false);
    }

    // spill t = 2*dot - sq_col to per-wave LDS (row-major, stride 20)
#pragma unroll
    for (int r = 0; r < 8; ++r)
      ct[(r + 8 * lh) * LDC + ll] = fmaf(2.0f, c[r], -sqc);
    asm volatile("s_wait_dscnt 0" ::: "memory");         // wave-local LDS ordering

    if (lane < 16) {
      const float* crow = &ct[lane * LDC];               // this lane's row (16B aligned)
#pragma unroll
      for (int jj = 0; jj < 16; jj += 4) {
        const v4f tq = *(const v4f*)(crow + jj);         // ds_load_b128: 4 candidates
        const float m = fmaxf(fmaxf(tq.x, tq.y), fmaxf(tq.z, tq.w));
        if (m >= kv[KNN - 1]) {                          // vectorized guard
#pragma unroll
          for (int u = 0; u < 4; ++u) {
            const float t  = tq[u];
            const int   cj = colBase + jj + u;
            if (t > kv[KNN - 1] || (t == kv[KNN - 1] && cj < ki[KNN - 1])) {
              float cv = t; int ci = cj;                 // bubble-insert (stable ties)
#pragma unroll
              for (int p = 0; p < KNN; ++p) {
                const bool b = (cv > kv[p]) || (cv == kv[p] && ci < ki[p]);
                const float tv = b ? kv[p] : cv; const int ti = b ? ki[p] : ci;
                kv[p] = b ? cv : kv[p];          ki[p] = b ? ci : ki[p];
                cv = tv; ci = ti;
              }
            }
          }
        }
      }
    }
    asm volatile("s_wait_dscnt 0" ::: "memory");
  }

  __syncthreads();                         // all waves done with sC tile regions
  if (lane < 16) {
#pragma unroll
    for (int s = 0; s < KNN; ++s) {
      sC [wave * 160 + lane * KNN + s] = kv[s];
      sMI[wave * 160 + lane * KNN + s] = ki[s];
    }
  }
  __syncthreads();

  if (tid < 16) {                          // merge 8 sorted lists per row
    const int row = rowBase + tid;
    float fv[KNN]; int fi[KNN];
#pragma unroll
    for (int s = 0; s < KNN; ++s) { fv[s] = -3.0e38f; fi[s] = 0x7fffffff; }
    for (int w = 0; w < 8; ++w)
      for (int s = 0; s < KNN; ++s) {
        float cv = sC[w * 160 + tid * KNN + s];
        int   ci = sMI[w * 160 + tid * KNN + s];
        if (cv > fv[KNN - 1] || (cv == fv[KNN - 1] && ci < fi[KNN - 1])) {
#pragma unroll
          for (int p = 0; p < KNN; ++p) {
            const bool b = (cv > fv[p]) || (cv == fv[p] && ci < fi[p]);
            const float tv = b ? fv[p] : cv; const int ti = b ? fi[p] : ci;
            fv[p] = b ? cv : fv[p];          fi[p] = b ? ci : fi[p];
            cv = tv; ci = ti;
          }
        }
      }
    const float si = sq[row];
#pragma unroll
    for (int s = 0; s < KNN; ++s) {
      tkv[row * KNN + s] = expf(0.5f * (fv[s] - si));    // back to A-values
      tki[row * KNN + s] = fi[s];
    }
  }
}

// ---------------------------------------------------------------------------
// Kernel 3: degree sums of gram = 0.5*(res + res^T), diag excluded
// ---------------------------------------------------------------------------
__global__ void degree_kernel(const float* __restrict__ tkv,
                              const int* __restrict__ tki, float* __restrict__ Dsum) {
  const int t = blockIdx.x * 256 + threadIdx.x;
  if (t >= NTOK * KNN) return;
  const int i = t / KNN, j = tki[t];
  if (j == i) return;
  const float hv = 0.5f * tkv[t];
  atomicAdd(&Dsum[i], hv);
  atomicAdd(&Dsum[j], hv);
}

// Kernel 4: D = Dsum>0 ? rsqrt(Dsum) : 0   (in place)
__global__ void rsqrt_kernel(float* __restrict__ Dsum) {
  const int i = blockIdx.x * 256 + threadIdx.x;
  if (i < NTOK) { const float s = Dsum[i]; Dsum[i] = s > 0.f ? rsqrtf(s) : 0.f; }
}

// ---------------------------------------------------------------------------
// Kernel 5: sparse gPsi = (D res_sym D) @ Psi ; one block per row, thread=col
// ---------------------------------------------------------------------------
__global__ __launch_bounds__(256) void gpsi_kernel(
    const float* __restrict__ Psi, const float* __restrict__ tkv,
    const int* __restrict__ tki, const float* __restrict__ D,
    float* __restrict__ gPsi) {
  const int i = blockIdx.x, c = threadIdx.x;
  const float Di    = D[i];
  const float psi_i = Psi[(size_t)i * KEIG + c];
  float acc = 0.f;
  for (int k = 0; k < KNN; ++k) {
    const int j = tki[i * KNN + k];
    if (j == i) continue;
    const float w = 0.5f * tkv[i * KNN + k] * Di * D[j];
    acc += w * Psi[(size_t)j * KEIG + c];
    atomicAdd(&gPsi[(size_t)j * KEIG + c], w * psi_i);
  }
  atomicAdd(&gPsi[(size_t)i * KEIG + c], acc);
}

// ---------------------------------------------------------------------------
// Kernel 6: R = Psi^T @ gPsi  [256x256], fp32 WMMA, one wave per 16x16 tile
// ---------------------------------------------------------------------------
__global__ __launch_bounds__(32) void r_wmma_kernel(
    const float* __restrict__ Psi, const float* __restrict__ gPsi,
    float* __restrict__ R) {
  const int tm = blockIdx.x >> 4, tn = blockIdx.x & 15;
  const int lane = threadIdx.x & 31, lh = lane >> 4, ll = lane & 15;
  const float* pA = Psi  + (size_t)(2 * lh) * KEIG + tm * 16 + ll;  // A[m,k]=Psi[k,m]
  const float* pB = gPsi + (size_t)(2 * lh) * KEIG + tn * 16 + ll;  // B[k,n]=gPsi[k,n]
  v8f c = {};
#pragma unroll 4
  for (int kk = 0; kk < NTOK; kk += 4) {
    v2f a, b;
    a.x = pA[kk * KEIG]; a.y = pA[kk * KEIG + KEIG];
    b.x = pB[kk * KEIG]; b.y = pB[kk * KEIG + KEIG];
    c = __builtin_amdgcn_wmma_f32_16x16x4_f32(false, a, false, b,
                                              (short)0, c, false, false);
  }
#pragma unroll
  for (int r = 0; r < 8; ++r)
    R[(size_t)(tm * 16 + r + 8 * lh) * KEIG + tn * 16 + ll] = c[r];
}

// ---------------------------------------------------------------------------
// Kernel 7: loss = -s^2 tr(R);  reg = alpha * s^4 * sum_{i<j} R_ij^2
// ---------------------------------------------------------------------------
__global__ __launch_bounds__(256) void finalize_kernel(
    const float* __restrict__ R, float* __restrict__ out) {
  __shared__ float sT[256], sU[256];
  const int r = threadIdx.x;
  float up = 0.f;
  for (int j = r + 1; j < KEIG; ++j) { const float v = R[r * KEIG + j]; up += v * v; }
  sT[r] = R[r * KEIG + r]; sU[r] = up;
  __syncthreads();
  for (int s = 128; s > 0; s >>= 1) {
    if (r < s) { sT[r] += sT[r + s]; sU[r] += sU[r + s]; }
    __syncthreads();
  }
  if (r == 0) {
    const float s2 = (10.0f / 8192.0f) * (10.0f / 8192.0f);  // (T/N)^2
    out[0] = -s2 * sT[0];
    out[1] = 0.05f * s2 * s2 * sU[0];                        // alpha * s^4 * triu
  }
}

// ---------------------------------------------------------------------------
extern "C" void kernel_launch(void* const* d_in, const int* in_sizes, int n_in,
                              void* d_out, int out_size, void* d_ws, size_t ws_size,
                              hipStream_t stream) {
  const float* FF  = (const float*)d_in[0];   // [8192,768]
  const float* Psi = (const float*)d_in[1];   // [8192,256]
  float* out = (float*)d_out;                 // [2]

  // workspace layout (~34.6 MB)
  float* FFn  = (float*)d_ws;                       // 8192*768
  float* sq   = FFn + (size_t)NTOK * DFEAT;         // 8192
  float* tkv  = sq + NTOK;                          // 8192*10
  int*   tki  = (int*)(tkv + NTOK * KNN);           // 8192*10
  float* Dsum = (float*)(tki + NTOK * KNN);         // 8192   (becomes D)
  float* gPsi = Dsum + NTOK;                        // 8192*256
  float* R    = gPsi + (size_t)NTOK * KEIG;         // 256*256

  // zero the accumulation region (Dsum | gPsi | R are contiguous)
  hipMemsetAsync(Dsum, 0, (size_t)(NTOK + NTOK * KEIG + KEIG * KEIG) * sizeof(float),
                 stream);

  normalize_kernel<<<NTOK / 8, 256, 0, stream>>>(FF, FFn, sq);
  sim_topk_kernel<<<NTOK / 16, 256, 0, stream>>>(FFn, sq, tkv, tki);
  degree_kernel<<<(NTOK * KNN + 255) / 256, 256, 0, stream>>>(tkv, tki, Dsum);
  rsqrt_kernel<<<NTOK / 256, 256, 0, stream>>>(Dsum);
  gpsi_kernel<<<NTOK, 256, 0, stream>>>(Psi, tkv, tki, Dsum, gPsi);
  r_wmma_kernel<<<256, 32, 0, stream>>>(Psi, gPsi, R);
  finalize_kernel<<<1, 256, 0, stream>>>(R, out);
}